// visual_loss_987842478713
// MI455X (gfx1250) — compile-verified
//
#include <hip/hip_runtime.h>

typedef float v2f __attribute__((ext_vector_type(2)));
typedef float v8f __attribute__((ext_vector_type(8)));

#define HDIM     100
#define WDIM     130
#define HW       (HDIM * WDIM)     // 13000
#define GTHREADS 256                // threads per frame-group (8 waves, wave32)
#define THREADS  (2 * GTHREADS)     // one block handles both frames
#define NWAVES   (THREADS / 32)     // 16

// Single fused kernel. The reference's pairwise sum telescopes exactly:
//   sum(r[:-1]) - sum(r[1:]) == r[0] - r[B-1]
// so only frames 0 and B-1 (channels 0 and team_no) are ever read: 208 KB
// instead of 320 MB. Waves 0-7 reduce frame 0, waves 8-15 reduce frame B-1.
// Per-frame reductions run through V_WMMA_F32_16X16X4_F32 (fp32 in, fp32 RNE
// accumulate): each wave folds 64 values/step into the 16x4 A operand against
// an all-ones B, so D[m][n] += sum_k A[m][k] independent of B's lane layout.
__global__ __launch_bounds__(THREADS)
void visual_loss_kernel(const float* __restrict__ x,
                        const int*  __restrict__ team_no_p,
                        float*      __restrict__ out,
                        int nframes)
{
    __shared__ float wts[WDIM];
    __shared__ float redq[NWAVES];
    __shared__ float redp[NWAVES];

    const int tid = threadIdx.x;

    // Triangular column weights: w_j = 65 - | -65 + j*(130/129) |
    if (tid < WDIM) {
        float xs = -65.0f + (float)tid * (130.0f / 129.0f);
        wts[tid] = 65.0f - fabsf(xs);
    }
    __syncthreads();

    const int team  = *team_no_p;
    const int group = tid >> 8;                     // 0: frame 0, 1: frame B-1
    const int gt    = tid & (GTHREADS - 1);
    const int frame = (group == 0) ? 0 : (nframes - 1);
    const float* __restrict__ ch0 = x + (size_t)frame * 3u * HW;
    const float* __restrict__ cht = ch0 + (size_t)team * HW;

    v8f cq = {0.f, 0.f, 0.f, 0.f, 0.f, 0.f, 0.f, 0.f};  // q = s*w*(1+xt)
    v8f cp = {0.f, 0.f, 0.f, 0.f, 0.f, 0.f, 0.f, 0.f};  // p = s   (gate)
    const v2f ones = {1.0f, 1.0f};

    // Each thread handles 2 adjacent pixels per iteration via one b64 load
    // per channel. 512 pixels per group-iteration -> 26 iterations.
    const int iters = (HW + 2 * GTHREADS - 1) / (2 * GTHREADS);  // 26
    for (int i = 0; i < iters; ++i) {
        int pa = i * (2 * GTHREADS) + 2 * gt;       // first pixel of the pair
        int pc = (pa < HW - 2) ? pa : (HW - 2);     // clamp so b64 stays in-bounds
        float m0 = (pa     < HW) ? 1.0f : 0.0f;
        float m1 = (pa + 1 < HW) ? 1.0f : 0.0f;

        float2 x0 = *reinterpret_cast<const float2*>(ch0 + pc);  // global_load_b64
        float2 xt = *reinterpret_cast<const float2*>(cht + pc);

        int   c0 = pc % WDIM;
        int   c1 = (c0 == WDIM - 1) ? 0 : (c0 + 1);
        float w0 = wts[c0];
        float w1 = wts[c1];

        // sigmoid via v_exp_f32 + v_rcp_f32 (no IEEE division chain)
        float s0 = __builtin_amdgcn_rcpf(1.0f + __expf(-x0.x));
        float s1 = __builtin_amdgcn_rcpf(1.0f + __expf(-x0.y));

        v2f aq = { s0 * w0 * (1.0f + xt.x) * m0,
                   s1 * w1 * (1.0f + xt.y) * m1 };
        v2f ap = { s0 * m0,
                   s1 * m1 };

        // 8 args: (neg_a, A, neg_b, B, c_mod, C, reuse_a, reuse_b)
        cq = __builtin_amdgcn_wmma_f32_16x16x4_f32(false, aq, false, ones,
                                                   (short)0, cq, false, false);
        cp = __builtin_amdgcn_wmma_f32_16x16x4_f32(false, ap, false, ones,
                                                   (short)0, cp, false, false);
    }

    // Per-lane: sum of 8 C VGPRs = D[0..7][col] (lanes 0-15) or D[8..15][col]
    // (lanes 16-31); pairing lane L with L^16 completes a full column, which
    // equals the grand total (all columns identical with B = ones).
    float lq = cq[0] + cq[1] + cq[2] + cq[3] + cq[4] + cq[5] + cq[6] + cq[7];
    float lp = cp[0] + cp[1] + cp[2] + cp[3] + cp[4] + cp[5] + cp[6] + cp[7];
    lq += __shfl_xor(lq, 16, 32);
    lp += __shfl_xor(lp, 16, 32);

    const int lane = tid & 31;
    const int wv   = tid >> 5;                      // 0..15; 0-7 frame0, 8-15 frameB-1
    if (lane == 0) { redq[wv] = lq; redp[wv] = lp; }
    __syncthreads();

    if (tid == 0) {
        float q0 = 0.f, p0 = 0.f, q1 = 0.f, p1 = 0.f;
        for (int w = 0; w < NWAVES / 2; ++w)      { q0 += redq[w]; p0 += redp[w]; }
        for (int w = NWAVES / 2; w < NWAVES; ++w) { q1 += redq[w]; p1 += redp[w]; }
        const float thresh = (1.0f / 6.0f) * (float)HDIM * (float)WDIM;  // 2166.67
        float r0 = (p0 <= thresh) ? q0 : 0.0f;
        float r1 = (p1 <= thresh) ? q1 : 0.0f;
        float diff = r0 - r1;                        // r[0] - r[B-1]
        out[0] = (diff != 0.0f) ? diff : -0.001f;
    }
}

extern "C" void kernel_launch(void* const* d_in, const int* in_sizes, int n_in,
                              void* d_out, int out_size, void* d_ws, size_t ws_size,
                              hipStream_t stream) {
    const float* x    = (const float*)d_in[0];
    const int*   team = (const int*)d_in[1];
    float*       out  = (float*)d_out;

    const int nframes = in_sizes[0] / (3 * HW);  // 2048

    visual_loss_kernel<<<1, THREADS, 0, stream>>>(x, team, out, nframes);
}